// Embedding_40072044872155
// MI455X (gfx1250) — compile-verified
//
#include <hip/hip_runtime.h>
#include <math.h>

typedef __attribute__((ext_vector_type(2))) float v2f;
typedef __attribute__((ext_vector_type(8))) float v8f;

#define HGT 512
#define WID 512
#define NSTATE (HGT * WID)        // 262144
#define DIN (3 * NSTATE)          // 786432
#define NSWEEP 10
#define NWAVES 4096
#define CHUNK (DIN / NWAVES)      // 192 = 3 * 64

// workspace offsets in floats
#define OFF_VA   0
#define OFF_VB   (NSTATE)
#define OFF_MSG  (2 * NSTATE)
#define OFF_FEAT (3 * NSTATE)                 // [v | r_out | r_in], 3*NSTATE
#define OFF_P1   (6 * NSTATE)                 // NWAVES*32 partials (W1 gemv)
#define OFF_P2   (6 * NSTATE + NWAVES * 32)   // NWAVES*32 partials (Wp1 gemv)
#define OFF_H2   (6 * NSTATE + 2 * NWAVES * 32)
#define OFF_POS  (OFF_H2 + 64)

__global__ void zero_va_kernel(float* __restrict__ va) {
  int i = blockIdx.x * 256 + threadIdx.x;
  if (i < NSTATE) va[i] = 0.0f;
}

// first-index argmax over x channel 1 (single workgroup)
__global__ void argmax_kernel(const float* __restrict__ x1, int* __restrict__ pos) {
  __shared__ float sv[256];
  __shared__ int   si[256];
  int t = threadIdx.x;
  float best = -INFINITY; int bi = 0;
  for (int k = t; k < NSTATE; k += 256) {
    float v = x1[k];
    if (v > best) { best = v; bi = k; }   // strictly > keeps lowest index per thread
  }
  sv[t] = best; si[t] = bi;
  __syncthreads();
  for (int s = 128; s > 0; s >>= 1) {
    if (t < s) {
      if (sv[t + s] > sv[t] || (sv[t + s] == sv[t] && si[t + s] < si[t])) {
        sv[t] = sv[t + s]; si[t] = si[t + s];
      }
    }
    __syncthreads();
  }
  if (t == 0) *pos = si[0];
}

// out[0:32] = W(32 x DIN) @ vec : split-K over NWAVES waves, WMMA f32 16x16x4.
// B is broadcast across all 16 columns, so every C column equals the GEMV result.
// K-permutation per 8-float chunk: wmma step A covers data K {k,k+1,k+4,k+5},
// step B covers {k+2,k+3,k+6,k+7}; A and B agree slot-by-slot, and each
// lane-half then streams contiguous float4s.
__global__ void gemv32_wmma_kernel(const float* __restrict__ Wm,
                                   const float* __restrict__ vec,
                                   float* __restrict__ partials) {
  const int lane = threadIdx.x & 31;
  const int half = lane >> 4;
  const int lr   = lane & 15;
  const int wave = __builtin_amdgcn_readfirstlane(
      blockIdx.x * 8 + (int)(threadIdx.x >> 5));   // scalar, 0..NWAVES-1
  const long long row0 = (long long)lr * DIN;
  const long long row1 = (long long)(lr + 16) * DIN;
  const int k0 = wave * CHUNK;
  const int hofs = 4 * half;
  v8f c0 = {};
  v8f c1 = {};
  for (int kb = k0; kb < k0 + CHUNK; kb += 64) {   // 3 iterations
    __builtin_prefetch(Wm + row0 + kb + hofs + 64, 0, 1);  // next 256B of each stream
    __builtin_prefetch(Wm + row1 + kb + hofs + 64, 0, 1);
#pragma unroll
    for (int kk = 0; kk < 64; kk += 8) {
      const int ks = kb + kk + hofs;
      const float4 tb = *(const float4*)(vec + ks);
      const float4 t0 = *(const float4*)(Wm + row0 + ks);
      const float4 t1 = *(const float4*)(Wm + row1 + ks);
      v2f bA; bA[0] = tb.x; bA[1] = tb.y;
      v2f bB; bB[0] = tb.z; bB[1] = tb.w;
      v2f a0A; a0A[0] = t0.x; a0A[1] = t0.y;
      v2f a0B; a0B[0] = t0.z; a0B[1] = t0.w;
      v2f a1A; a1A[0] = t1.x; a1A[1] = t1.y;
      v2f a1B; a1B[0] = t1.z; a1B[1] = t1.w;
      c0 = __builtin_amdgcn_wmma_f32_16x16x4_f32(false, a0A, false, bA, (short)0, c0, false, false);
      c1 = __builtin_amdgcn_wmma_f32_16x16x4_f32(false, a1A, false, bA, (short)0, c1, false, false);
      c0 = __builtin_amdgcn_wmma_f32_16x16x4_f32(false, a0B, false, bB, (short)0, c0, false, false);
      c1 = __builtin_amdgcn_wmma_f32_16x16x4_f32(false, a1B, false, bB, (short)0, c1, false, false);
    }
  }
  // lanes 0 and 16 hold rows 0..7 / 8..15 of each tile (any column works: broadcast B)
  if (lr == 0) {
    float* p = partials + wave * 32;
    const int rb = half * 8;
#pragma unroll
    for (int r = 0; r < 8; ++r) {
      p[rb + r]      = c0[r];
      p[16 + rb + r] = c1[r];
    }
  }
}

// h1 = relu(sum(partials)+b1); h2 = relu(W2@h1+b2). One block, 64 threads.
__global__ void h12_kernel(const float* __restrict__ partials,
                           const float* __restrict__ b1,
                           const float* __restrict__ W2,
                           const float* __restrict__ b2,
                           float* __restrict__ h2out) {
  __shared__ float h1s[32];
  int t = threadIdx.x;
  if (t < 32) {
    float s = b1[t];
    for (int w = 0; w < NWAVES; ++w) s += partials[w * 32 + t];
    h1s[t] = fmaxf(s, 0.0f);
  }
  __syncthreads();
  float s = b2[t];
  for (int j = 0; j < 32; ++j) s += W2[t * 32 + j] * h1s[j];
  h2out[t] = fmaxf(s, 0.0f);
}

// r_out / r_in: 262144x64 GEMV each via WMMA. One 16-row tile per wave.
__global__ void rori_wmma_kernel(const float* __restrict__ Wro,
                                 const float* __restrict__ bro,
                                 const float* __restrict__ Wri,
                                 const float* __restrict__ bri,
                                 const float* __restrict__ h2,
                                 float* __restrict__ feat) {
  const int tile = __builtin_amdgcn_readfirstlane(
      blockIdx.x * 8 + (int)(threadIdx.x >> 5));      // scalar, 0..32767
  const int second = (tile >= (NSTATE / 16)) ? 1 : 0;
  const int r0 = (tile & (NSTATE / 16 - 1)) * 16;
  const float* Wm = second ? Wri : Wro;
  const float* bb = second ? bri : bro;
  float* out = feat + NSTATE + second * NSTATE;
  const int lane = threadIdx.x & 31;
  const int half = lane >> 4;
  const int lr   = lane & 15;
  const int hofs = 4 * half;
  const float* arow = Wm + (long long)(r0 + lr) * 64;
  v8f c = {};
#pragma unroll
  for (int k = 0; k < 64; k += 8) {
    const int ks = k + hofs;
    const float4 tb = *(const float4*)(h2 + ks);
    const float4 ta = *(const float4*)(arow + ks);
    v2f bA; bA[0] = tb.x; bA[1] = tb.y;
    v2f bB; bB[0] = tb.z; bB[1] = tb.w;
    v2f aA; aA[0] = ta.x; aA[1] = ta.y;
    v2f aB; aB[0] = ta.z; aB[1] = ta.w;
    c = __builtin_amdgcn_wmma_f32_16x16x4_f32(false, aA, false, bA, (short)0, c, false, false);
    c = __builtin_amdgcn_wmma_f32_16x16x4_f32(false, aB, false, bB, (short)0, c, false, false);
  }
  if (lr == 0) {
    const int rb = r0 + half * 8;
#pragma unroll
    for (int r = 0; r < 8; ++r) out[rb + r] = c[r] + bb[rb + r];
  }
}

__global__ void msg_kernel(const float* __restrict__ feat, float* __restrict__ msg) {
  int i = blockIdx.x * 256 + threadIdx.x;
  if (i < NSTATE) msg[i] = feat[2 * NSTATE + i] - feat[NSTATE + i];
}

// one value-iteration sweep: dst = max(src, max over 4-neighbours of (src+msg))
__global__ void sweep_kernel(const float* __restrict__ src,
                             const float* __restrict__ msg,
                             float* __restrict__ dst) {
  int idx = blockIdx.x * 256 + threadIdx.x;
  if (idx >= NSTATE) return;
  int i = idx >> 9, j = idx & 511;
  float cand = -INFINITY;
  if (i > 0)       cand = fmaxf(cand, src[idx - 512] + msg[idx - 512]);
  if (i < HGT - 1) cand = fmaxf(cand, src[idx + 512] + msg[idx + 512]);
  if (j > 0)       cand = fmaxf(cand, src[idx - 1]   + msg[idx - 1]);
  if (j < WID - 1) cand = fmaxf(cand, src[idx + 1]   + msg[idx + 1]);
  dst[idx] = fmaxf(src[idx], cand);
}

// p1 = sum(partials2)+bp1 (no relu); logits = Wph@p1+bph; softmax; value pick.
__global__ void logits_kernel(const float* __restrict__ partials2,
                              const float* __restrict__ bp1,
                              const float* __restrict__ Wph,
                              const float* __restrict__ bph,
                              const float* __restrict__ feat,
                              const int* __restrict__ pos,
                              float* __restrict__ out) {
  __shared__ float p1[32];
  __shared__ float lg[4];
  int t = threadIdx.x;
  if (t < 32) {
    float s = bp1[t];
    for (int w = 0; w < NWAVES; ++w) s += partials2[w * 32 + t];
    p1[t] = s;
  }
  __syncthreads();
  if (t < 4) {
    float s = bph[t];
    for (int j = 0; j < 32; ++j) s += Wph[t * 32 + j] * p1[j];
    lg[t] = s;
  }
  __syncthreads();
  if (t == 0) {
    float m = fmaxf(fmaxf(lg[0], lg[1]), fmaxf(lg[2], lg[3]));
    float e0 = expf(lg[0] - m), e1 = expf(lg[1] - m);
    float e2 = expf(lg[2] - m), e3 = expf(lg[3] - m);
    float inv = 1.0f / (e0 + e1 + e2 + e3);
    out[0] = e0 * inv; out[1] = e1 * inv; out[2] = e2 * inv; out[3] = e3 * inv;
    out[4] = feat[*pos];   // state value v[pi,pj]
  }
}

extern "C" void kernel_launch(void* const* d_in, const int* in_sizes, int n_in,
                              void* d_out, int out_size, void* d_ws, size_t ws_size,
                              hipStream_t stream) {
  const float* x   = (const float*)d_in[0];
  const float* W1  = (const float*)d_in[1];
  const float* b1  = (const float*)d_in[2];
  const float* W2  = (const float*)d_in[3];
  const float* b2  = (const float*)d_in[4];
  const float* Wro = (const float*)d_in[5];
  const float* bro = (const float*)d_in[6];
  const float* Wri = (const float*)d_in[7];
  const float* bri = (const float*)d_in[8];
  const float* Wp1 = (const float*)d_in[9];
  const float* bp1 = (const float*)d_in[10];
  const float* Wph = (const float*)d_in[11];
  const float* bph = (const float*)d_in[12];
  float* ws = (float*)d_ws;
  float* out = (float*)d_out;

  float* va   = ws + OFF_VA;
  float* vb   = ws + OFF_VB;
  float* msg  = ws + OFF_MSG;
  float* feat = ws + OFF_FEAT;
  float* p1   = ws + OFF_P1;
  float* p2   = ws + OFF_P2;
  float* h2   = ws + OFF_H2;
  int*   pos  = (int*)(ws + OFF_POS);

  const int nb = NSTATE / 256;  // 1024

  zero_va_kernel<<<nb, 256, 0, stream>>>(va);
  argmax_kernel<<<1, 256, 0, stream>>>(x + NSTATE, pos);

  // h1 partials: W1 @ xf
  gemv32_wmma_kernel<<<NWAVES / 8, 256, 0, stream>>>(W1, x, p1);
  h12_kernel<<<1, 64, 0, stream>>>(p1, b1, W2, b2, h2);

  // r_out, r_in into feat[NSTATE:3*NSTATE]
  rori_wmma_kernel<<<(2 * NSTATE / 16) / 8, 256, 0, stream>>>(Wro, bro, Wri, bri, h2, feat);
  msg_kernel<<<nb, 256, 0, stream>>>(feat, msg);

  // 10 sweeps, ping-pong; final sweep writes v into feat[0:NSTATE]
  for (int s = 0; s < NSWEEP; ++s) {
    const float* src = (s % 2 == 0) ? va : vb;
    float* dst = (s == NSWEEP - 1) ? feat : ((s % 2 == 0) ? vb : va);
    sweep_kernel<<<nb, 256, 0, stream>>>(src, msg, dst);
  }

  // p1 partials: Wp1 @ feat
  gemv32_wmma_kernel<<<NWAVES / 8, 256, 0, stream>>>(Wp1, feat, p2);
  logits_kernel<<<1, 32, 0, stream>>>(p2, bp1, Wph, bph, feat, pos, out);

  (void)in_sizes; (void)n_in; (void)out_size; (void)ws_size;
}